// GCN_text_61959198212218
// MI455X (gfx1250) — compile-verified
//
#include <hip/hip_runtime.h>
#include <hip/hip_bf16.h>

typedef float v2f __attribute__((ext_vector_type(2)));
typedef float v8f __attribute__((ext_vector_type(8)));

#define IN_C  128
#define OUT_C 128

// ---------------------------------------------------------------------------
// 1) degree init (self loop contributes 1)
// ---------------------------------------------------------------------------
__global__ void k_deg_init(float* __restrict__ deg, int n) {
  int i = blockIdx.x * blockDim.x + threadIdx.x;
  if (i < n) deg[i] = 1.0f;
}

// 2) in-degree count over edge destinations
__global__ void k_deg_count(const long long* __restrict__ dst, int E,
                            float* __restrict__ deg) {
  int e = blockIdx.x * blockDim.x + threadIdx.x;
  if (e < E) atomicAdd(&deg[(int)dst[e]], 1.0f);
}

// 3) dinv = rsqrt(deg)
__global__ void k_rsqrt(const float* __restrict__ deg, float* __restrict__ dinv,
                        int n) {
  int i = blockIdx.x * blockDim.x + threadIdx.x;
  if (i < n) dinv[i] = rsqrtf(deg[i]);
}

// ---------------------------------------------------------------------------
// 4) GEMM  h = x @ W   via V_WMMA_F32_16X16X4_F32 (exact fp32 matrix path)
//    block = 256 threads = 8 waves; wave w owns output tile (16 rows) x
//    (cols 16w..16w+15); grid.x = N/16 row tiles. W staged in LDS (padded).
// ---------------------------------------------------------------------------
#define LDSW 132  // padded row stride (floats) to avoid half-wave bank alias

__global__ __launch_bounds__(256)
void k_gemm_wmma(const float* __restrict__ x, const float* __restrict__ W,
                 float* __restrict__ h) {
  __shared__ float lW[IN_C * LDSW];

  // cooperative stage of full 128x128 W into LDS
  for (int i = threadIdx.x; i < IN_C * OUT_C; i += 256) {
    int k = i >> 7, n = i & 127;
    lW[k * LDSW + n] = W[i];
  }
  __syncthreads();

  const int wave = threadIdx.x >> 5;       // 0..7 -> column tile
  const int lane = threadIdx.x & 31;
  const int m    = lane & 15;              // row within 16-row tile
  const int hi   = lane >> 4;              // 0: K{0,1}/M0-7 half, 1: K{2,3}/M8-15
  const int koff = hi << 1;                // 0 or 2
  const int row0 = blockIdx.x * 16;
  const int col  = wave * 16 + m;          // output column (also B column)

  const float* xrow = x + (size_t)(row0 + m) * IN_C;

  v8f c = {};
#pragma unroll
  for (int kk = 0; kk < IN_C; kk += 4) {
    // A 16x4 fp32 fragment: VGPR0 = K=kk+koff, VGPR1 = K=kk+koff+1
    v2f a;
    a.x = xrow[kk + koff];
    a.y = xrow[kk + koff + 1];
    // B 4x16 fp32 fragment (mirrored striping): lane holds column `col`
    v2f b;
    b.x = lW[(kk + koff)     * LDSW + col];
    b.y = lW[(kk + koff + 1) * LDSW + col];
    c = __builtin_amdgcn_wmma_f32_16x16x4_f32(
        /*neg_a=*/false, a, /*neg_b=*/false, b,
        /*c_mod=*/(short)0, c, /*reuse_a=*/false, /*reuse_b=*/false);
  }

  // D 16x16 f32: VGPR r -> row (r + 8*hi), col = lane&15 within tile
  float* hp = h + (size_t)(row0 + hi * 8) * OUT_C + col;
#pragma unroll
  for (int r = 0; r < 8; ++r) hp[(size_t)r * OUT_C] = c[r];
}

// ---------------------------------------------------------------------------
// 5) out = h * dinv^2  (self-loop term; full overwrite -> replay-deterministic)
// ---------------------------------------------------------------------------
__global__ void k_self_loop(const float* __restrict__ h,
                            const float* __restrict__ dinv,
                            float* __restrict__ out, size_t total) {
  size_t i = (size_t)blockIdx.x * blockDim.x + threadIdx.x;
  if (i < total) {
    float w = dinv[i >> 7];
    out[i] = h[i] * (w * w);
  }
}

// ---------------------------------------------------------------------------
// 6) edge scatter: one wave per edge, float4 gather + fp32 global atomics
// ---------------------------------------------------------------------------
__global__ __launch_bounds__(256)
void k_edge_scatter(const long long* __restrict__ src,
                    const long long* __restrict__ dst,
                    const float* __restrict__ h,
                    const float* __restrict__ dinv,
                    float* __restrict__ out, int E) {
  int e = blockIdx.x * 8 + (threadIdx.x >> 5);
  if (e >= E) return;
  const int lane = threadIdx.x & 31;
  const int s = (int)src[e];
  const int d = (int)dst[e];
  const float w = dinv[s] * dinv[d];

  const float4 v = ((const float4*)(h + (size_t)s * OUT_C))[lane];
  float* op = out + (size_t)d * OUT_C + lane * 4;
  atomicAdd(op + 0, v.x * w);
  atomicAdd(op + 1, v.y * w);
  atomicAdd(op + 2, v.z * w);
  atomicAdd(op + 3, v.w * w);
}

// ---------------------------------------------------------------------------
// 7) out = PReLU(out + bias)
// ---------------------------------------------------------------------------
__global__ void k_bias_prelu(float* __restrict__ out,
                             const float* __restrict__ bias,
                             const float* __restrict__ alpha, size_t total) {
  size_t i = (size_t)blockIdx.x * blockDim.x + threadIdx.x;
  if (i < total) {
    float v = out[i] + bias[i & 127];
    float a = alpha[0];
    out[i] = (v >= 0.0f) ? v : a * v;
  }
}

// ---------------------------------------------------------------------------
extern "C" void kernel_launch(void* const* d_in, const int* in_sizes, int n_in,
                              void* d_out, int out_size, void* d_ws, size_t ws_size,
                              hipStream_t stream) {
  const float*     x     = (const float*)d_in[0];
  const long long* eidx  = (const long long*)d_in[1];
  const float*     W     = (const float*)d_in[2];
  const float*     bias  = (const float*)d_in[3];
  const float*     alpha = (const float*)d_in[4];

  const int N = in_sizes[0] / IN_C;
  const int E = in_sizes[1] / 2;
  const long long* src = eidx;
  const long long* dst = eidx + E;

  // workspace layout: deg[N] | dinv[N] | h[N*OUT_C]
  float* deg  = (float*)d_ws;
  float* dinv = deg + N;
  float* h    = dinv + N;

  float* out = (float*)d_out;
  const size_t total = (size_t)N * OUT_C;

  k_deg_init  <<<(N + 255) / 256, 256, 0, stream>>>(deg, N);
  k_deg_count <<<(E + 255) / 256, 256, 0, stream>>>(dst, E, deg);
  k_rsqrt     <<<(N + 255) / 256, 256, 0, stream>>>(deg, dinv, N);

  k_gemm_wmma <<<N / 16, 256, 0, stream>>>(x, W, h);

  k_self_loop <<<(int)((total + 255) / 256), 256, 0, stream>>>(h, dinv, out, total);
  k_edge_scatter <<<(E + 7) / 8, 256, 0, stream>>>(src, dst, h, dinv, out, E);
  k_bias_prelu<<<(int)((total + 255) / 256), 256, 0, stream>>>(out, bias, alpha, total);
}